// RGATLayer_34351148434173
// MI455X (gfx1250) — compile-verified
//
#include <hip/hip_runtime.h>
#include <hip/hip_bf16.h>
#include <stdint.h>

// Problem constants (from reference)
#define B_  4
#define S_  256
#define H_  128
#define NR_ 40
#define NEG_INF_ (-1e18f)

// ---------------- gfx1250 hardware tanh (transcendental unit) ----------------
__device__ __forceinline__ float tanh_hw(float x) {
#if __has_builtin(__builtin_amdgcn_tanhf)
  return __builtin_amdgcn_tanhf(x);
#elif __has_builtin(__builtin_amdgcn_tanh_f32)
  return __builtin_amdgcn_tanh_f32(x);
#else
  float r;
  asm("v_tanh_f32 %0, %1" : "=v"(r) : "v"(x));
  return r;
#endif
}

// LDS byte offset of a shared-memory object (generic ptr low 32 bits == LDS offset)
__device__ __forceinline__ uint32_t lds_off32(const void* p) {
  return (uint32_t)(uintptr_t)p;
}

// ---------------- CDNA5 async global->LDS (ASYNCcnt path) -------------------
__device__ __forceinline__ void async_ld_b128(uint32_t lds_byte_off,
                                              const float* sbase,
                                              uint32_t global_byte_off) {
  asm volatile("global_load_async_to_lds_b128 %0, %1, %2 offset:0"
               :
               : "v"(lds_byte_off), "v"(global_byte_off), "s"(sbase)
               : "memory");
}

__device__ __forceinline__ void async_ld_b32(uint32_t lds_byte_off,
                                             const int* sbase,
                                             uint32_t global_byte_off) {
  asm volatile("global_load_async_to_lds_b32 %0, %1, %2 offset:0"
               :
               : "v"(lds_byte_off), "v"(global_byte_off), "s"(sbase)
               : "memory");
}

__device__ __forceinline__ void wait_asynccnt0() {
#if __has_builtin(__builtin_amdgcn_s_wait_asynccnt)
  __builtin_amdgcn_s_wait_asynccnt(0);
#else
  asm volatile("s_wait_asynccnt 0x0" ::: "memory");
#endif
}

// ---------------- CDNA5 Tensor Data Mover (TENSORcnt path) ------------------
#if __has_builtin(__builtin_amdgcn_tensor_load_to_lds) && \
    __has_builtin(__builtin_amdgcn_s_wait_tensorcnt)
#define HAVE_TDM_ 1
typedef uint32_t v4u_ __attribute__((ext_vector_type(4)));
typedef int      v8i_ __attribute__((ext_vector_type(8)));
typedef int      v4i_ __attribute__((ext_vector_type(4)));

// DMA one NR_*H_ f32 table (20 KB, contiguous) global -> LDS with a single
// TDM descriptor: 1D tensor of N elements, one N x 1 tile, data_size=4B.
// D# bit layout per CDNA5 ISA 8.3/8.4. This toolchain exposes the 6-arg
// builtin: (g0 v4u, g1 v8i, g2 v4i, g3 v4i, v8i, i32 cpol).
__device__ __forceinline__ void tdm_load_table(const float* gsrc, float* ldst) {
  const uint64_t ga = (uint64_t)(uintptr_t)gsrc;
  const uint32_t la = lds_off32(ldst);
  const uint32_t N  = NR_ * H_;   // 5120 elements
  // Group0: count=1 | lds_addr[63:32] | global_addr[120:64] | type=2 [127:126]
  v4u_ g0 = { 1u,
              la,
              (uint32_t)ga,
              (uint32_t)((ga >> 32) & 0x01FFFFFFu) | (2u << 30) };
  // Group1: wg_mask=0, data_size=2(4B) [17:16], tensor_dim0=N [79:48],
  //         tensor_dim1=1 [111:80], tile_dim0=N [127:112], tile_dim1=1 [143:128],
  //         tile_dim2=0, tensor_dim0_stride=N [207:160], tensor_dim1_stride=N.
  v8i_ g1 = { (int)(2u << 16),
              (int)((N & 0xFFFFu) << 16),
              (int)(((N >> 16) & 0xFFFFu) | (1u << 16)),
              (int)((N & 0xFFFFu) << 16),
              1,
              (int)N,
              (int)((N & 0xFFFFu) << 16),
              0 };
  v4i_ gz4 = { 0, 0, 0, 0 };              // groups 2/3: 2D tile, unused
  v8i_ gz8 = { 0, 0, 0, 0, 0, 0, 0, 0 };  // trailing group: unused
  __builtin_amdgcn_tensor_load_to_lds(g0, g1, gz4, gz4, gz8, 0);
}
#else
#define HAVE_TDM_ 0
#endif

__global__ __launch_bounds__(256) void rgat_rowattn_kernel(
    const float* __restrict__ query, const float* __restrict__ key,
    const float* __restrict__ value, const float* __restrict__ rel_emb,
    const float* __restrict__ qw_emb, const float* __restrict__ kw_emb,
    const float* __restrict__ vw_emb, const int* __restrict__ src_len,
    const int* __restrict__ rel_idx, float* __restrict__ out)
{
  // LDS: four 40x128 f32 tables (80 KB) + row scratch (~3.6 KB); gfx1250 WGP
  // LDS is 320 KB so ~3 workgroups co-reside.
  __shared__ __align__(16) float s_rel[NR_ * H_];
  __shared__ __align__(16) float s_qw [NR_ * H_];
  __shared__ __align__(16) float s_kw [NR_ * H_];
  __shared__ __align__(16) float s_vw [NR_ * H_];
  __shared__ float s_sc [S_];     // scores -> attention weights
  __shared__ float s_acc[H_];     // phase-2 half-sum combine
  __shared__ float s_red[16];     // block reductions
  __shared__ int   s_r1 [S_];     // rel_idx[b,k,q]
  __shared__ int   s_r2 [S_];     // rel_idx[b,q,k]

  const int tid  = threadIdx.x;
  const int lane = tid & 31;
  const int wav  = tid >> 5;          // 8 waves (wave32)
  const int b    = blockIdx.x / S_;
  const int q    = blockIdx.x % S_;

  // ---- Stage embedding tables into LDS ----
#if HAVE_TDM_
  // One TDM DMA per table, issued once by wave 0; other 7 waves skip staging.
  if (wav == 0) {
    tdm_load_table(rel_emb, s_rel);
    tdm_load_table(qw_emb,  s_qw);
    tdm_load_table(kw_emb,  s_kw);
    tdm_load_table(vw_emb,  s_vw);
  }
#else
  // Fallback: per-lane async b128 DMAs (5 per thread per table).
  {
    const float* srcs[4] = { rel_emb, qw_emb, kw_emb, vw_emb };
    float*       dsts[4] = { s_rel,   s_qw,   s_kw,   s_vw   };
#pragma unroll
    for (int t = 0; t < 4; ++t) {
      const uint32_t base = lds_off32(dsts[t]);
#pragma unroll
      for (int i = 0; i < 5; ++i) {
        const uint32_t off = (uint32_t)(tid + i * 256) * 16u;
        async_ld_b128(base + off, srcs[t], off);
      }
    }
  }
#endif

  // ---- Gather relation ids for this row via async global->LDS b32 ----
  // s_r1[k] = rel_idx[b,k,q] (strided), s_r2[k] = rel_idx[b,q,k] (contiguous)
  async_ld_b32(lds_off32(s_r1) + (uint32_t)tid * 4u, rel_idx,
               (uint32_t)(((b * S_ + tid) * S_ + q) * 4));
  async_ld_b32(lds_off32(s_r2) + (uint32_t)tid * 4u, rel_idx,
               (uint32_t)(((b * S_ + q) * S_ + tid) * 4));

  // Per-lane key[b,q,h] registers, h = lane + 32*j (coalesced per wave).
  float kq[4];
#pragma unroll
  for (int j = 0; j < 4; ++j)
    kq[j] = key[(size_t)(b * S_ + q) * H_ + (lane + 32 * j)];

  wait_asynccnt0();                        // this wave's async DMAs landed
#if HAVE_TDM_
  if (wav == 0) __builtin_amdgcn_s_wait_tensorcnt(0);  // TDM tiles landed
#endif
  __syncthreads();                         // all staging visible to all waves

  // ---- Phase 1: s[k] = sum_h rel[r1,h]*tanh(qw[r1,h]*Q[k,h] + kw[r2,h]*K[q,h])
  for (int kb = 0; kb < S_; kb += 8) {
    const int   k    = kb + wav;                 // one k per wave
    const int   r1   = s_r1[k];
    const int   r2   = s_r2[k];
    const float* qrw = query + (size_t)(b * S_ + k) * H_;
    float part = 0.f;
#pragma unroll
    for (int j = 0; j < 4; ++j) {
      const int h = lane + 32 * j;               // conflict-free LDS banks
      const float a = s_qw[r1 * H_ + h] * qrw[h] + s_kw[r2 * H_ + h] * kq[j];
      part += s_rel[r1 * H_ + h] * tanh_hw(a);
    }
#pragma unroll
    for (int off = 16; off > 0; off >>= 1)       // wave32 shuffle reduce
      part += __shfl_xor(part, off, 32);
    if (lane == 0) s_sc[k] = part;
  }
  __syncthreads();

  // ---- Masked softmax over k (256 elements, one per thread) ----
  const int sl = src_len[b];
  const float v = (tid < sl) ? s_sc[tid] : NEG_INF_;

  float m = v;
#pragma unroll
  for (int off = 16; off > 0; off >>= 1) m = fmaxf(m, __shfl_xor(m, off, 32));
  if (lane == 0) s_red[wav] = m;
  __syncthreads();
  if (wav == 0) {
    float t = (lane < 8) ? s_red[lane] : NEG_INF_;
#pragma unroll
    for (int off = 4; off > 0; off >>= 1) t = fmaxf(t, __shfl_xor(t, off, 32));
    if (lane == 0) s_red[8] = t;
  }
  __syncthreads();
  const float mx = s_red[8];

  const float e = (tid < sl) ? __expf(v - mx) : 0.f;
  float sum = e;
#pragma unroll
  for (int off = 16; off > 0; off >>= 1) sum += __shfl_xor(sum, off, 32);
  if (lane == 0) s_red[wav] = sum;
  __syncthreads();
  if (wav == 0) {
    float t = (lane < 8) ? s_red[lane] : 0.f;
#pragma unroll
    for (int off = 4; off > 0; off >>= 1) t += __shfl_xor(t, off, 32);
    if (lane == 0) s_red[9] = t;
  }
  __syncthreads();
  s_sc[tid] = e * (1.0f / s_red[9]);             // normalized attention
  __syncthreads();

  // ---- Phase 2: out[q,h] = sum_k att[k] * vw[r1(k),h] * V[k,h] ----
  const int g = tid >> 7;                        // two k-halves
  const int h = tid & (H_ - 1);
  const int k0 = g * (S_ / 2);
  float acc = 0.f;
#pragma unroll 8
  for (int kk = 0; kk < S_ / 2; ++kk) {
    const int k = k0 + kk;
    const float a = s_sc[k];
    const int  r1 = s_r1[k];
    acc += a * s_vw[r1 * H_ + h] * value[(size_t)(b * S_ + k) * H_ + h];
  }
  if (g == 1) s_acc[h] = acc;
  __syncthreads();
  if (g == 0) out[(size_t)(b * S_ + q) * H_ + h] = acc + s_acc[h];
}

extern "C" void kernel_launch(void* const* d_in, const int* in_sizes, int n_in,
                              void* d_out, int out_size, void* d_ws, size_t ws_size,
                              hipStream_t stream) {
  (void)in_sizes; (void)n_in; (void)out_size; (void)d_ws; (void)ws_size;
  const float* query   = (const float*)d_in[0];
  const float* key     = (const float*)d_in[1];
  const float* value   = (const float*)d_in[2];
  const float* rel_emb = (const float*)d_in[3];
  const float* qw_emb  = (const float*)d_in[4];
  const float* kw_emb  = (const float*)d_in[5];
  const float* vw_emb  = (const float*)d_in[6];
  const int*   src_len = (const int*)d_in[7];
  const int*   rel_idx = (const int*)d_in[8];
  float*       out     = (float*)d_out;

  dim3 grid(B_ * S_);     // one workgroup per (b, q) row
  dim3 block(256);        // 8 wave32 waves
  rgat_rowattn_kernel<<<grid, block, 0, stream>>>(
      query, key, value, rel_emb, qw_emb, kw_emb, vw_emb, src_len, rel_idx, out);
}